// MOM_81810537055456
// MI455X (gfx1250) — compile-verified
//
#include <hip/hip_runtime.h>
#include <math.h>
#include <stdint.h>

typedef __attribute__((ext_vector_type(2))) float v2f;
typedef __attribute__((ext_vector_type(8))) float v8f;

#define CB 2        // batch
#define CS 2048     // seq
#define CD 256      // model dim
#define CH 8        // heads
#define CM 8        // memories
#define CHD 2048    // H*D

#if __has_builtin(__builtin_amdgcn_global_load_async_to_lds_b128)
#define MOM_HAVE_ASYNC 1
#endif

// exact pointee type the builtin wants: 16-byte int vector
typedef int mom_v4i __attribute__((vector_size(4 * sizeof(int))));
typedef __attribute__((address_space(1))) mom_v4i* mom_g128p;
typedef __attribute__((address_space(3))) mom_v4i* mom_l128p;

// 16-byte global->LDS copy: async on gfx1250, sync staging fallback otherwise.
__device__ __forceinline__ void mom_cp16(const float* g, float* l) {
#ifdef MOM_HAVE_ASYNC
  __builtin_amdgcn_global_load_async_to_lds_b128(
      (mom_g128p)(uintptr_t)g, (mom_l128p)(uintptr_t)l, 0, 0);
#else
  float4 v = *(const float4*)g;
  *(float4*)l = v;
#endif
}

#ifdef MOM_HAVE_ASYNC
#if __has_builtin(__builtin_amdgcn_s_wait_asynccnt)
#define MOM_WAIT_ASYNC(n) __builtin_amdgcn_s_wait_asynccnt(n)
#else
#define MOM_WAIT_ASYNC(n) asm volatile("s_wait_asynccnt %0" ::"n"(n) : "memory")
#endif
#else
#define MOM_WAIT_ASYNC(n)
#endif

__device__ __forceinline__ v8f mom_wmma4(v2f a, v2f b, v8f c) {
  // D = A(16x4, f32) * B(4x16, f32) + C(16x16, f32)
  return __builtin_amdgcn_wmma_f32_16x16x4_f32(false, a, false, b, (short)0, c,
                                               false, false);
}

// ---------------------------------------------------------------------------
// 1. Router: scores = softmax(x@Wr+br); biased top-2 (first-index ties like
//    lax.top_k); routing = renormalized masked softmax scores.  [B*S, M]
// ---------------------------------------------------------------------------
__global__ void mom_routing_kernel(const float* __restrict__ x,
                                   const float* __restrict__ Wr,
                                   const float* __restrict__ br,
                                   const float* __restrict__ bias0,
                                   float* __restrict__ routing) {
  int t = blockIdx.x * blockDim.x + threadIdx.x;
  if (t >= CB * CS) return;
  int b = t / CS;
  float sc[CM];
#pragma unroll
  for (int j = 0; j < CM; ++j) sc[j] = br[j];
  const float* xt = x + (long)t * CD;
  for (int d = 0; d < CD; ++d) {
    float xv = xt[d];
#pragma unroll
    for (int j = 0; j < CM; ++j) sc[j] += xv * Wr[d * CM + j];
  }
  float mx = sc[0];
#pragma unroll
  for (int j = 1; j < CM; ++j) mx = fmaxf(mx, sc[j]);
  float sum = 0.f;
#pragma unroll
  for (int j = 0; j < CM; ++j) { sc[j] = expf(sc[j] - mx); sum += sc[j]; }
  float inv = 1.f / sum;
#pragma unroll
  for (int j = 0; j < CM; ++j) sc[j] *= inv;
  float best1 = -1e30f; int i1 = 0;
#pragma unroll
  for (int j = 0; j < CM; ++j) {
    float bj = sc[j] + bias0[b * CM + j];
    if (bj > best1) { best1 = bj; i1 = j; }
  }
  float best2 = -1e30f; int i2 = (i1 == 0) ? 1 : 0;
#pragma unroll
  for (int j = 0; j < CM; ++j) {
    if (j == i1) continue;
    float bj = sc[j] + bias0[b * CM + j];
    if (bj > best2) { best2 = bj; i2 = j; }
  }
  float invn = 1.f / (sc[i1] + sc[i2]);
#pragma unroll
  for (int j = 0; j < CM; ++j)
    routing[(long)t * CM + j] = (j == i1 || j == i2) ? sc[j] * invn : 0.f;
}

// ---------------------------------------------------------------------------
// 2. First moments: g1[b,m,:] = sum_s r*x ; g0[b,m] = sum_s r
// ---------------------------------------------------------------------------
__global__ void mom_g1_kernel(const float* __restrict__ x,
                              const float* __restrict__ routing,
                              float* __restrict__ g1, float* __restrict__ g0) {
  int bm = blockIdx.x;
  int d = threadIdx.x;
  int b = bm >> 3, m = bm & 7;
  const float* xb = x + (long)b * CS * CD;
  const float* rb = routing + (long)b * CS * CM;
  float acc = 0.f, sr = 0.f;
  for (int s = 0; s < CS; ++s) {
    float r = rb[s * CM + m];
    acc += r * xb[(long)s * CD + d];
    sr += r;
  }
  g1[bm * CD + d] = acc;
  if (d == 0) g0[bm] = sr;
}

#define UB_KC 32
#define UB_AS 36  // padded LDS row stride for U-kernel A panel

// ---------------------------------------------------------------------------
// 3. G[b,m] = X^T diag(r_m) X  ([256,2048-K]) — 128x64 block tile, BK=32,
//    double-buffered async global->LDS x panels (A staged [k][i], B [k][j]),
//    routing scalars applied by VALU at fragment build (lane-uniform loads).
// ---------------------------------------------------------------------------
__global__ __launch_bounds__(256) void mom_G_kernel(
    const float* __restrict__ x, const float* __restrict__ routing,
    float* __restrict__ G) {
  __shared__ __align__(16) float sA[2][UB_KC * 128];  // [k][i]
  __shared__ __align__(16) float sB[2][UB_KC * 64];   // [k][j]
  int blk = blockIdx.x;            // bm*8 + bi*4 + bj
  int bm = blk >> 3;
  int bi = (blk >> 2) & 1;
  int bj = blk & 3;
  int b = bm >> 3, m = bm & 7;
  int i0b = bi << 7;               // 0 or 128
  int j0b = bj << 6;               // 0..192
  const float* xb = x + (long)b * CS * CD;
  const float* rb = routing + (long)b * CS * CM;
  int tid = threadIdx.x;
  int w = tid >> 5, lane = tid & 31, lo = lane & 15, hi = lane >> 4;
  int rw = w << 4;

  int eA = tid * 16;
  int rA = eA >> 7, cA = eA & 127;  // A panel: 32 rows x 128 cols
  int eB = tid * 8;
  int rB = eB >> 6, cB = eB & 63;

  // prologue: panel 0
  {
#pragma unroll
    for (int c = 0; c < 16; c += 4)
      mom_cp16(xb + (long)rA * CD + i0b + cA + c, &sA[0][rA * 128 + cA + c]);
#pragma unroll
    for (int c = 0; c < 8; c += 4)
      mom_cp16(xb + (long)rB * CD + j0b + cB + c, &sB[0][rB * 64 + cB + c]);
  }
  v8f acc[4] = {};
  const int NP = CS / UB_KC;  // 64 panels
  for (int p = 0; p < NP; ++p) {
    int cur = p & 1;
    if (p < NP - 1) {
      int kn = (p + 1) * UB_KC;
      int nxt = cur ^ 1;
#pragma unroll
      for (int c = 0; c < 16; c += 4)
        mom_cp16(xb + (long)(kn + rA) * CD + i0b + cA + c,
                 &sA[nxt][rA * 128 + cA + c]);
#pragma unroll
      for (int c = 0; c < 8; c += 4)
        mom_cp16(xb + (long)(kn + rB) * CD + j0b + cB + c,
                 &sB[nxt][rB * 64 + cB + c]);
      MOM_WAIT_ASYNC(6);
    } else {
      MOM_WAIT_ASYNC(0);
    }
    __syncthreads();
    const float* Al = sA[cur];
    const float* Bl = sB[cur];
    int kbase = p * UB_KC;
#pragma unroll
    for (int k0l = 0; k0l < UB_KC; k0l += 4) {
      int ka = k0l + 2 * hi;
      float r0 = rb[(kbase + ka + 0) * CM + m];   // lane-uniform broadcast
      float r1 = rb[(kbase + ka + 1) * CM + m];
      v2f a;                                      // A[i,k] = r[k]*x[k,i]
      a[0] = r0 * Al[(ka + 0) * 128 + rw + lo];
      a[1] = r1 * Al[(ka + 1) * 128 + rw + lo];
#pragma unroll
      for (int t = 0; t < 4; ++t) {
        v2f bb;
        bb[0] = Bl[(ka + 0) * 64 + t * 16 + lo];
        bb[1] = Bl[(ka + 1) * 64 + t * 16 + lo];
        acc[t] = mom_wmma4(a, bb, acc[t]);
      }
    }
    __syncthreads();
  }
  float* Gt = G + (long)bm * CD * CD;
#pragma unroll
  for (int t = 0; t < 4; ++t)
#pragma unroll
    for (int v = 0; v < 8; ++v)
      Gt[(i0b + rw + v + 8 * hi) * CD + j0b + t * 16 + lo] = acc[t][v];
}

// ---------------------------------------------------------------------------
// 4. U[b,m] = G[b,m] @ Wv[m]  ([256,256]@[256,2048])
//    128x64 block tile, BK=32, double-buffered async global->LDS panels,
//    8 waves x (16x64, 4 accumulators).
// ---------------------------------------------------------------------------
__global__ __launch_bounds__(256) void mom_U_kernel(
    const float* __restrict__ G, const float* __restrict__ Wv,
    float* __restrict__ U) {
  __shared__ __align__(16) float sA[2][128 * UB_AS];
  __shared__ __align__(16) float sB[2][UB_KC * 64];
  int blk = blockIdx.x;            // bm*64 + bi*32 + bj
  int bm = blk >> 6;
  int bi = (blk >> 5) & 1;
  int bj = blk & 31;
  int m = bm & 7;
  int r0 = bi << 7;                // 0 or 128
  int j0b = bj << 6;               // 0..1984
  const float* Gt = G + (long)bm * CD * CD;
  const float* Wvm = Wv + (long)m * CD * CHD;
  int tid = threadIdx.x;
  int w = tid >> 5, lane = tid & 31, lo = lane & 15, hi = lane >> 4;
  int rw = w << 4;                 // wave's 16 rows within 128

  // panel copy helper indices (per thread: A = 4x b128, B = 2x b128)
  int eA = tid * 16;
  int rA = eA >> 5, cA = eA & 31;  // 16 elems never cross a 32-col row
  int eB = tid * 8;
  int rB = eB >> 6, cB = eB & 63;

  // prologue: panel 0
  {
#pragma unroll
    for (int c = 0; c < 16; c += 4)
      mom_cp16(Gt + (long)(r0 + rA) * CD + 0 + cA + c,
               &sA[0][rA * UB_AS + cA + c]);
#pragma unroll
    for (int c = 0; c < 8; c += 4)
      mom_cp16(Wvm + (long)(0 + rB) * CHD + j0b + cB + c,
               &sB[0][rB * 64 + cB + c]);
  }
  v8f acc[4] = {};
  for (int kt = 0; kt < CD / UB_KC; ++kt) {
    int cur = kt & 1;
    if (kt < CD / UB_KC - 1) {
      int kn = (kt + 1) * UB_KC;
      int nxt = cur ^ 1;
#pragma unroll
      for (int c = 0; c < 16; c += 4)
        mom_cp16(Gt + (long)(r0 + rA) * CD + kn + cA + c,
                 &sA[nxt][rA * UB_AS + cA + c]);
#pragma unroll
      for (int c = 0; c < 8; c += 4)
        mom_cp16(Wvm + (long)(kn + rB) * CHD + j0b + cB + c,
                 &sB[nxt][rB * 64 + cB + c]);
      MOM_WAIT_ASYNC(6);
    } else {
      MOM_WAIT_ASYNC(0);
    }
    __syncthreads();
    const float* Al = sA[cur];
    const float* Bl = sB[cur];
#pragma unroll
    for (int k0l = 0; k0l < UB_KC; k0l += 4) {
      int ka = k0l + 2 * hi;
      v2f a;
      a[0] = Al[(rw + lo) * UB_AS + ka];
      a[1] = Al[(rw + lo) * UB_AS + ka + 1];
#pragma unroll
      for (int t = 0; t < 4; ++t) {
        v2f bb;
        bb[0] = Bl[ka * 64 + t * 16 + lo];
        bb[1] = Bl[(ka + 1) * 64 + t * 16 + lo];
        acc[t] = mom_wmma4(a, bb, acc[t]);
      }
    }
    __syncthreads();
  }
  float* Ut = U + (long)bm * CD * CHD;
#pragma unroll
  for (int t = 0; t < 4; ++t)
#pragma unroll
    for (int v = 0; v < 8; ++v)
      Ut[(long)(r0 + rw + v + 8 * hi) * CHD + j0b + t * 16 + lo] = acc[t][v];
}

// ---------------------------------------------------------------------------
// 5. mem[b,h] = sum_m Wk[m][:,hblk]^T @ U[b,m][:,hblk]
//    Same async double-buffered structure; A panel staged transposed in LDS
//    as [kk][i] (contiguous i) so WMMA A-fragment reads are conflict-free.
// ---------------------------------------------------------------------------
__global__ __launch_bounds__(256) void mom_mem_kernel(
    const float* __restrict__ Wk, const float* __restrict__ U,
    float* __restrict__ mem) {
  __shared__ __align__(16) float sA[2][UB_KC * 128];  // [kk][i]
  __shared__ __align__(16) float sB[2][UB_KC * 64];   // [kk][j]
  int blk = blockIdx.x;            // bh*8 + bi*4 + bj
  int bh = blk >> 3;
  int bi = (blk >> 2) & 1;
  int bj = blk & 3;
  int b = bh >> 3, h = bh & 7;
  int i0b = bi << 7;
  int j0b = bj << 6;
  int tid = threadIdx.x;
  int w = tid >> 5, lane = tid & 31, lo = lane & 15, hi = lane >> 4;
  int rw = w << 4;

  int eA = tid * 16;
  int rA = eA >> 7, cA = eA & 127;  // A panel: 32 rows x 128 cols
  int eB = tid * 8;
  int rB = eB >> 6, cB = eB & 63;

  const float* WkB = Wk + h * CD;           // column block h of Wk[m]
  const float* UB = U + h * CD;             // column block h of U[b,m]

  // prologue: panel (m=0, kt=0)
  {
    const float* Wkm = WkB;                          // m = 0
    const float* Ut = UB + (long)(b * CM) * CD * CHD;
#pragma unroll
    for (int c = 0; c < 16; c += 4)
      mom_cp16(Wkm + (long)rA * CHD + i0b + cA + c, &sA[0][rA * 128 + cA + c]);
#pragma unroll
    for (int c = 0; c < 8; c += 4)
      mom_cp16(Ut + (long)rB * CHD + j0b + cB + c, &sB[0][rB * 64 + cB + c]);
  }
  v8f acc[4] = {};
  const int NP = CM * (CD / UB_KC);  // 64 panels
  for (int p = 0; p < NP; ++p) {
    int cur = p & 1;
    if (p < NP - 1) {
      int pn = p + 1;
      int mn = pn >> 3, kn = (pn & 7) * UB_KC;
      int nxt = cur ^ 1;
      const float* Wkm = WkB + (long)mn * CD * CHD;
      const float* Ut = UB + (long)(b * CM + mn) * CD * CHD;
#pragma unroll
      for (int c = 0; c < 16; c += 4)
        mom_cp16(Wkm + (long)(kn + rA) * CHD + i0b + cA + c,
                 &sA[nxt][rA * 128 + cA + c]);
#pragma unroll
      for (int c = 0; c < 8; c += 4)
        mom_cp16(Ut + (long)(kn + rB) * CHD + j0b + cB + c,
                 &sB[nxt][rB * 64 + cB + c]);
      MOM_WAIT_ASYNC(6);
    } else {
      MOM_WAIT_ASYNC(0);
    }
    __syncthreads();
    const float* Al = sA[cur];
    const float* Bl = sB[cur];
#pragma unroll
    for (int k0l = 0; k0l < UB_KC; k0l += 4) {
      int ka = k0l + 2 * hi;
      v2f a;                                     // A[i,k] = Wk[kk, hD+i]
      a[0] = Al[(ka + 0) * 128 + rw + lo];
      a[1] = Al[(ka + 1) * 128 + rw + lo];
#pragma unroll
      for (int t = 0; t < 4; ++t) {
        v2f bb;
        bb[0] = Bl[(ka + 0) * 64 + t * 16 + lo];
        bb[1] = Bl[(ka + 1) * 64 + t * 16 + lo];
        acc[t] = mom_wmma4(a, bb, acc[t]);
      }
    }
    __syncthreads();
  }
  float* mt = mem + (long)bh * CD * CD;
#pragma unroll
  for (int t = 0; t < 4; ++t)
#pragma unroll
    for (int v = 0; v < 8; ++v)
      mt[(i0b + rw + v + 8 * hi) * CD + j0b + t * 16 + lo] = acc[t][v];
}

// ---------------------------------------------------------------------------
// 6a. t1[b,m] = Wk[m]^T g1[b,m] ; t2[b,m] = Wv[m]^T g1[b,m]
// ---------------------------------------------------------------------------
__global__ void mom_t12_kernel(const float* __restrict__ Wk,
                               const float* __restrict__ Wv,
                               const float* __restrict__ g1,
                               float* __restrict__ t1, float* __restrict__ t2) {
  int bm = blockIdx.x >> 3;
  int col = ((blockIdx.x & 7) << 8) + threadIdx.x;
  int m = bm & 7;
  const float* g = g1 + bm * CD;
  float a1 = 0.f, a2 = 0.f;
  for (int k = 0; k < CD; ++k) {
    float gv = g[k];
    a1 += gv * Wk[((long)m * CD + k) * CHD + col];
    a2 += gv * Wv[((long)m * CD + k) * CHD + col];
  }
  t1[(long)bm * CHD + col] = a1;
  t2[(long)bm * CHD + col] = a2;
}

// 6b. exact rank-1 bias corrections.
__global__ void mom_membias_kernel(const float* __restrict__ t1,
                                   const float* __restrict__ t2,
                                   const float* __restrict__ g0,
                                   const float* __restrict__ bk,
                                   const float* __restrict__ bv,
                                   float* __restrict__ mem) {
  long idx = (long)blockIdx.x * blockDim.x + threadIdx.x;
  int j = idx & 255;
  int i = (idx >> 8) & 255;
  int h = (idx >> 16) & 7;
  int b = (int)(idx >> 19);
  float add = 0.f;
#pragma unroll
  for (int m = 0; m < CM; ++m) {
    int bm = b * CM + m;
    float t1v = t1[(long)bm * CHD + h * CD + i];
    float bvv = bv[(long)m * CHD + h * CD + j];
    float bkv = bk[(long)m * CHD + h * CD + i];
    float t2v = t2[(long)bm * CHD + h * CD + j];
    add += t1v * bvv + bkv * (t2v + g0[bm] * bvv);
  }
  mem[idx] += add;
}

// ---------------------------------------------------------------------------
// 7. P[b,h] = mem[b,h] @ Wo[hblk,:]  — 16x64 wave tiles
// ---------------------------------------------------------------------------
__global__ __launch_bounds__(128) void mom_P_kernel(
    const float* __restrict__ mem, const float* __restrict__ Wo,
    float* __restrict__ P) {
  int wave = blockIdx.x * (blockDim.x >> 5) + (threadIdx.x >> 5);
  if (wave >= CB * CH * 16 * 4) return;
  int jg = wave & 3;
  int it = (wave >> 2) & 15;
  int bh = wave >> 6;
  int h = bh & 7;
  int i0 = it << 4, j0 = jg << 6;
  int lane = threadIdx.x & 31, lo = lane & 15, hi = lane >> 4;
  const float* mt = mem + (long)bh * CD * CD;
  const float* Woh = Wo + (long)h * CD * CD;
  v8f acc[4] = {};
  for (int k0 = 0; k0 < CD; k0 += 4) {
    int ka = k0 + 2 * hi;
    v2f a;
    a[0] = mt[(i0 + lo) * CD + ka + 0];
    a[1] = mt[(i0 + lo) * CD + ka + 1];
#pragma unroll
    for (int t = 0; t < 4; ++t) {
      v2f bb;
      bb[0] = Woh[(ka + 0) * CD + j0 + t * 16 + lo];
      bb[1] = Woh[(ka + 1) * CD + j0 + t * 16 + lo];
      acc[t] = mom_wmma4(a, bb, acc[t]);
    }
  }
  float* Pt = P + (long)bh * CD * CD;
#pragma unroll
  for (int t = 0; t < 4; ++t)
#pragma unroll
    for (int v = 0; v < 8; ++v)
      Pt[(i0 + v + 8 * hi) * CD + j0 + t * 16 + lo] = acc[t][v];
}

// ---------------------------------------------------------------------------
// 8. A[b] = sum_h Wq[:,hblk] @ P[b,h]  — 16x64 wave tiles, K = 2048
// ---------------------------------------------------------------------------
__global__ __launch_bounds__(128) void mom_A_kernel(
    const float* __restrict__ Wq, const float* __restrict__ P,
    float* __restrict__ Abuf) {
  int wave = blockIdx.x * (blockDim.x >> 5) + (threadIdx.x >> 5);
  if (wave >= CB * 16 * 4) return;
  int jg = wave & 3;
  int it = (wave >> 2) & 15;
  int b = wave >> 6;
  int i0 = it << 4, j0 = jg << 6;
  int lane = threadIdx.x & 31, lo = lane & 15, hi = lane >> 4;
  const float* Pb = P + (long)b * CH * CD * CD;
  v8f acc[4] = {};
  for (int k0 = 0; k0 < CHD; k0 += 4) {
    int ka = k0 + 2 * hi;
    v2f a;
    a[0] = Wq[(long)(i0 + lo) * CHD + ka + 0];
    a[1] = Wq[(long)(i0 + lo) * CHD + ka + 1];
    const float* P0 = Pb + (long)((ka + 0) >> 8) * CD * CD + ((ka + 0) & 255) * CD;
    const float* P1 = Pb + (long)((ka + 1) >> 8) * CD * CD + ((ka + 1) & 255) * CD;
#pragma unroll
    for (int t = 0; t < 4; ++t) {
      v2f bb;
      bb[0] = P0[j0 + t * 16 + lo];
      bb[1] = P1[j0 + t * 16 + lo];
      acc[t] = mom_wmma4(a, bb, acc[t]);
    }
  }
  float* At = Abuf + (long)b * CD * CD;
#pragma unroll
  for (int t = 0; t < 4; ++t)
#pragma unroll
    for (int v = 0; v < 8; ++v)
      At[(i0 + v + 8 * hi) * CD + j0 + t * 16 + lo] = acc[t][v];
}

// ---------------------------------------------------------------------------
// 9a. u[b,h,:] = bq_h^T mem[b,h]   9b. c[b,:] = sum_hv u*Wo + bo
// ---------------------------------------------------------------------------
__global__ void mom_u_kernel(const float* __restrict__ bq,
                             const float* __restrict__ mem,
                             float* __restrict__ u) {
  int idx = blockIdx.x * blockDim.x + threadIdx.x;
  int v = idx & 255;
  int h = (idx >> 8) & 7;
  int b = idx >> 11;
  const float* mt = mem + (long)(b * CH + h) * CD * CD;
  const float* bqh = bq + h * CD;
  float acc = 0.f;
  for (int k = 0; k < CD; ++k) acc += bqh[k] * mt[k * CD + v];
  u[idx] = acc;
}

__global__ void mom_c_kernel(const float* __restrict__ u,
                             const float* __restrict__ Wo,
                             const float* __restrict__ bo,
                             float* __restrict__ cvec) {
  int idx = blockIdx.x * blockDim.x + threadIdx.x;
  int e = idx & 255;
  int b = idx >> 8;
  float acc = bo[e];
  for (int hv = 0; hv < CHD; ++hv)
    acc += u[b * CHD + hv] * Wo[(long)hv * CD + e];
  cvec[idx] = acc;
}

// ---------------------------------------------------------------------------
// 10. out[b,s,:] = x[b,s,:] @ A[b] + c[b]  — 16x64 wave tiles
// ---------------------------------------------------------------------------
__global__ __launch_bounds__(128) void mom_out_kernel(
    const float* __restrict__ x, const float* __restrict__ Abuf,
    const float* __restrict__ cvec, float* __restrict__ out) {
  int wave = blockIdx.x * (blockDim.x >> 5) + (threadIdx.x >> 5);
  if (wave >= CB * 128 * 4) return;
  int jg = wave & 3;
  int st = (wave >> 2) & 127;
  int b = wave >> 9;
  int s0 = st << 4, j0 = jg << 6;
  int lane = threadIdx.x & 31, lo = lane & 15, hi = lane >> 4;
  const float* xb = x + (long)b * CS * CD;
  const float* At = Abuf + (long)b * CD * CD;
  v8f acc[4] = {};
  for (int k0 = 0; k0 < CD; k0 += 4) {
    int ka = k0 + 2 * hi;
    v2f a;
    a[0] = xb[(long)(s0 + lo) * CD + ka + 0];
    a[1] = xb[(long)(s0 + lo) * CD + ka + 1];
#pragma unroll
    for (int t = 0; t < 4; ++t) {
      v2f bb;
      bb[0] = At[(ka + 0) * CD + j0 + t * 16 + lo];
      bb[1] = At[(ka + 1) * CD + j0 + t * 16 + lo];
      acc[t] = mom_wmma4(a, bb, acc[t]);
    }
  }
  float* ob = out + (long)b * CS * CD;
#pragma unroll
  for (int t = 0; t < 4; ++t) {
    float cv = cvec[b * CD + j0 + t * 16 + lo];
#pragma unroll
    for (int v = 0; v < 8; ++v)
      ob[(long)(s0 + v + 8 * hi) * CD + j0 + t * 16 + lo] = acc[t][v] + cv;
  }
}

// ---------------------------------------------------------------------------
// Launch: all on `stream`, stream order = dependency order.
// ---------------------------------------------------------------------------
extern "C" void kernel_launch(void* const* d_in, const int* in_sizes, int n_in,
                              void* d_out, int out_size, void* d_ws,
                              size_t ws_size, hipStream_t stream) {
  const float* x = (const float*)d_in[0];
  const float* Wq = (const float*)d_in[1];
  const float* bq = (const float*)d_in[2];
  const float* Wr = (const float*)d_in[3];
  const float* br = (const float*)d_in[4];
  const float* Wk = (const float*)d_in[5];
  const float* bk = (const float*)d_in[6];
  const float* Wv = (const float*)d_in[7];
  const float* bv = (const float*)d_in[8];
  const float* Wo = (const float*)d_in[9];
  const float* bo = (const float*)d_in[10];
  const float* bias0 = (const float*)d_in[11];
  float* out = (float*)d_out;
  float* ws = (float*)d_ws;

  // workspace layout (floats); all offsets 16B-aligned
  float* routing = ws;                        // 32768
  float* g1 = ws + 32768;                     // 4096
  float* g0 = ws + 36864;                     // 16
  float* t1 = ws + 40960;                     // 32768
  float* t2 = ws + 73728;                     // 32768
  float* G = ws + 106496;                     // 1048576
  float* U = ws + 1155072;                    // 8388608
  float* mem = ws + 9543680;                  // 1048576
  float* P = ws + 10592256;                   // 1048576
  float* Abuf = ws + 11640832;                // 131072
  float* ubuf = ws + 11771904;                // 4096
  float* cvec = ws + 11776000;                // 512  -> total ~47.1 MB

  mom_routing_kernel<<<(CB * CS) / 256, 256, 0, stream>>>(x, Wr, br, bias0,
                                                          routing);
  mom_g1_kernel<<<CB * CM, CD, 0, stream>>>(x, routing, g1, g0);
  mom_G_kernel<<<CB * CM * 8, 256, 0, stream>>>(x, routing, G);
  mom_U_kernel<<<CB * CM * 64, 256, 0, stream>>>(G, Wv, U);
  mom_mem_kernel<<<CB * CH * 8, 256, 0, stream>>>(Wk, U, mem);
  mom_t12_kernel<<<CB * CM * 8, 256, 0, stream>>>(Wk, Wv, g1, t1, t2);
  mom_membias_kernel<<<(CB * CH * CD * CD) / 256, 256, 0, stream>>>(
      t1, t2, g0, bk, bv, mem);
  mom_P_kernel<<<(CB * CH * 16 * 4) / 4, 128, 0, stream>>>(mem, Wo, P);
  mom_A_kernel<<<(CB * 16 * 4) / 4, 128, 0, stream>>>(Wq, P, Abuf);
  mom_u_kernel<<<(CB * CH * CD) / 256, 256, 0, stream>>>(bq, mem, ubuf);
  mom_c_kernel<<<(CB * CD) / 256, 256, 0, stream>>>(ubuf, Wo, bo, cvec);
  mom_out_kernel<<<(CB * 128 * 4) / 4, 128, 0, stream>>>(x, Abuf, cvec, out);
}